// OntologyEmbedding_65987877535833
// MI455X (gfx1250) — compile-verified
//
#include <hip/hip_runtime.h>
#include <math.h>

#define N_NODES 100000
#define C 128
#define E_EDGES 800000
#define V_OUT 50000
#define NEG_SLOPE 0.2f
#define TOT_EDGES (E_EDGES + N_NODES)
#define N_STRIPS (N_NODES / 16)   // 6250, exact

typedef __attribute__((ext_vector_type(2))) float v2f;
typedef __attribute__((ext_vector_type(8))) float v8f;

// ---------------------------------------------------------------------------
// w_s = W^T @ att_src, w_d = W^T @ att_dst  (128-vectors; tiny, 1 block)
// ---------------------------------------------------------------------------
__global__ void prep_wsd(const float* __restrict__ W, const float* __restrict__ as,
                         const float* __restrict__ ad, float* __restrict__ ws,
                         float* __restrict__ wd) {
  int k = threadIdx.x;
  float s = 0.f, d = 0.f;
  for (int c = 0; c < C; ++c) {
    float w = W[c * C + k];
    s += as[c] * w;
    d += ad[c] * w;
  }
  ws[k] = s;
  wd[k] = d;
}

// out[n][c] = bias[c]; m[n] = -inf; denom[n] = 0
__global__ void init_layer(float* __restrict__ out, const float* __restrict__ bias,
                           float* __restrict__ m, float* __restrict__ den) {
  int i = blockIdx.x * blockDim.x + threadIdx.x;
  if (i < N_NODES * C) out[i] = bias[i & (C - 1)];
  if (i < N_NODES) { m[i] = -INFINITY; den[i] = 0.f; }
}

// ---------------------------------------------------------------------------
// h = x @ W^T via V_WMMA_F32_16X16X4_F32, fused a_s = x.w_s, a_d = x.w_d.
// One wave per 16-row strip; 8 waves (256 thr) per block; W^T staged in LDS.
// ---------------------------------------------------------------------------
__global__ __launch_bounds__(256)
void gemm_gat(const float* __restrict__ x, const float* __restrict__ W,
              const float* __restrict__ ws, const float* __restrict__ wd,
              float* __restrict__ h, float* __restrict__ a_s,
              float* __restrict__ a_d) {
  __shared__ float Wt[C * C];  // Wt[k*C + c] = W[c*C + k]  (B matrix, row-major in K)
  for (int i = threadIdx.x; i < C * C; i += 256) {
    int k = i >> 7, c = i & (C - 1);
    Wt[i] = W[c * C + k];      // LDS writes fully coalesced / conflict-free
  }
  __syncthreads();

  int wave = threadIdx.x >> 5;
  int lane = threadIdx.x & 31;
  int strip = blockIdx.x * 8 + wave;
  if (strip >= N_STRIPS) return;

  int row0 = strip * 16;
  int mrow = lane & 15;            // A: matrix row (M), B: matrix col (N)
  int koff = (lane >> 4) << 1;     // lanes 0-15 -> K{0,1}; lanes 16-31 -> K{2,3}
  const float* xrow = x + (size_t)(row0 + mrow) * C;

  v8f acc[8] = {};                 // 16 x 128 strip = 8 column tiles
  float asum = 0.f, adsum = 0.f;

  for (int k0 = 0; k0 < C; k0 += 4) {
    v2f a;
    a.x = xrow[k0 + koff];
    a.y = xrow[k0 + koff + 1];
    asum  += a.x * ws[k0 + koff] + a.y * ws[k0 + koff + 1];
    adsum += a.x * wd[k0 + koff] + a.y * wd[k0 + koff + 1];
#pragma unroll
    for (int t = 0; t < 8; ++t) {
      v2f b;
      b.x = Wt[(k0 + koff) * C + t * 16 + mrow];
      b.y = Wt[(k0 + koff + 1) * C + t * 16 + mrow];
      acc[t] = __builtin_amdgcn_wmma_f32_16x16x4_f32(
          false, a, false, b, (short)0, acc[t], false, false);
    }
  }

  // C/D layout: vgpr v, lanes 0-15 -> M=v, lanes 16-31 -> M=v+8; N = lane&15
  int rbase = row0 + ((lane >> 4) << 3);
#pragma unroll
  for (int t = 0; t < 8; ++t) {
    int col = t * 16 + mrow;
#pragma unroll
    for (int v = 0; v < 8; ++v)
      h[(size_t)(rbase + v) * C + col] = acc[t][v];
  }

  // lanes l and l^16 hold the two halves of row (row0 + mrow)'s K-dot
  asum  += __shfl_xor(asum, 16, 32);
  adsum += __shfl_xor(adsum, 16, 32);
  if (lane < 16) {
    a_s[row0 + mrow] = asum;
    a_d[row0 + mrow] = adsum;
  }
}

// ---------------------------------------------------------------------------
// Edge softmax (scatter)
// ---------------------------------------------------------------------------
__device__ __forceinline__ void atomicMaxF(float* addr, float v) {
  // monotonic-bit-pattern trick; m initialized to -inf
  if (v >= 0.f) atomicMax((int*)addr, __float_as_int(v));
  else          atomicMin((unsigned int*)addr, __float_as_uint(v));
}

__global__ void edge_scores(const int* __restrict__ edges, const float* __restrict__ a_s,
                            const float* __restrict__ a_d, float* __restrict__ ebuf,
                            float* __restrict__ m) {
  int i = blockIdx.x * blockDim.x + threadIdx.x;
  if (i >= TOT_EDGES) return;
  int s, d;
  if (i < E_EDGES) { s = edges[i]; d = edges[E_EDGES + i]; }
  else             { s = i - E_EDGES; d = s; }               // self loops
  float e = a_s[s] + a_d[d];
  e = (e >= 0.f) ? e : NEG_SLOPE * e;
  ebuf[i] = e;
  atomicMaxF(m + d, e);
}

__global__ void edge_exp(const int* __restrict__ edges, const float* __restrict__ m,
                         float* __restrict__ ebuf, float* __restrict__ den) {
  int i = blockIdx.x * blockDim.x + threadIdx.x;
  if (i >= TOT_EDGES) return;
  int d = (i < E_EDGES) ? edges[E_EDGES + i] : (i - E_EDGES);
  float ex = __expf(ebuf[i] - m[d]);
  ebuf[i] = ex;
  atomicAdd(den + d, ex);
}

// One wave per edge: lane handles 4 channels (float4 gather from L2-resident h,
// 4 f32 atomic adds into out[dst]).
__global__ __launch_bounds__(256)
void aggregate(const int* __restrict__ edges, const float* __restrict__ ex,
               const float* __restrict__ den, const float* __restrict__ h,
               float* __restrict__ out) {
  int eidx = blockIdx.x * 8 + (threadIdx.x >> 5);
  if (eidx >= TOT_EDGES) return;
  int lane = threadIdx.x & 31;
  int s, d;
  if (eidx < E_EDGES) { s = edges[eidx]; d = edges[E_EDGES + eidx]; }
  else                { s = eidx - E_EDGES; d = s; }
  float alpha = ex[eidx] / (den[d] + 1e-16f);
  float4 hv = ((const float4*)(h + (size_t)s * C))[lane];
  float* o = out + (size_t)d * C + lane * 4;
  atomicAdd(o + 0, alpha * hv.x);
  atomicAdd(o + 1, alpha * hv.y);
  atomicAdd(o + 2, alpha * hv.z);
  atomicAdd(o + 3, alpha * hv.w);
}

__global__ void gather_rows(const float* __restrict__ src, const int* __restrict__ idx,
                            float* __restrict__ dst) {
  int i = blockIdx.x * blockDim.x + threadIdx.x;
  if (i >= V_OUT * C) return;
  int v = i >> 7, c = i & (C - 1);
  dst[i] = src[(size_t)idx[v] * C + c];
}

// ---------------------------------------------------------------------------
extern "C" void kernel_launch(void* const* d_in, const int* in_sizes, int n_in,
                              void* d_out, int out_size, void* d_ws, size_t ws_size,
                              hipStream_t stream) {
  const float* emb = (const float*)d_in[0];
  const float* W1  = (const float*)d_in[1];
  const float* as1 = (const float*)d_in[2];
  const float* ad1 = (const float*)d_in[3];
  const float* b1  = (const float*)d_in[4];
  const float* W2  = (const float*)d_in[5];
  const float* as2 = (const float*)d_in[6];
  const float* ad2 = (const float*)d_in[7];
  const float* b2  = (const float*)d_in[8];
  const int* edges1 = (const int*)d_in[9];
  const int* edges2 = (const int*)d_in[10];
  const int* idxmap = (const int*)d_in[11];
  float* out = (float*)d_out;

  const size_t NC = (size_t)N_NODES * C;
  float* ws   = (float*)d_ws;
  float* h    = ws;                 // [N, C]
  float* outA = ws + NC;            // [N, C] layer1 out -> layer2 in -> layer2 out
  float* a_s  = ws + 2 * NC;        // [N]
  float* a_d  = a_s + N_NODES;      // [N]
  float* mbuf = a_d + N_NODES;      // [N]
  float* den  = mbuf + N_NODES;     // [N]
  float* ebuf = den + N_NODES;      // [E+N]
  float* wsv  = ebuf + TOT_EDGES;   // [C]
  float* wdv  = wsv + C;            // [C]

  const int gemm_blocks = (N_STRIPS + 7) / 8;
  const int edge_blocks = (TOT_EDGES + 255) / 256;
  const int agg_blocks  = (TOT_EDGES + 7) / 8;
  const int node_blocks = (int)((NC + 255) / 256);

  auto layer = [&](const float* x, const float* W, const float* as, const float* ad,
                   const float* bias, const int* edges, float* outl) {
    prep_wsd  <<<1, C, 0, stream>>>(W, as, ad, wsv, wdv);
    // GEMM reads x BEFORE init overwrites outl (allows x == outl aliasing, layer 2)
    gemm_gat  <<<gemm_blocks, 256, 0, stream>>>(x, W, wsv, wdv, h, a_s, a_d);
    init_layer<<<node_blocks, 256, 0, stream>>>(outl, bias, mbuf, den);
    edge_scores<<<edge_blocks, 256, 0, stream>>>(edges, a_s, a_d, ebuf, mbuf);
    edge_exp  <<<edge_blocks, 256, 0, stream>>>(edges, mbuf, ebuf, den);
    aggregate <<<agg_blocks, 256, 0, stream>>>(edges, ebuf, den, h, outl);
  };

  layer(emb,  W1, as1, ad1, b1, edges1, outA);   // layer 1: emb  -> outA
  layer(outA, W2, as2, ad2, b2, edges2, outA);   // layer 2: outA -> outA (safe)

  gather_rows<<<(V_OUT * C + 255) / 256, 256, 0, stream>>>(outA, idxmap, out);
}